// Efficient8BitALU_Bitwise_7945689497932
// MI455X (gfx1250) — compile-verified
//
#include <hip/hip_runtime.h>

// BD row layout
#define BD        100
#define ROWP      104     // padded LDS row (keeps float4 alignment, breaks bank stride)
#define RPW       8       // rows per wave (16 (pos,p) pairs -> one 16x16 WMMA tile)
#define WPB       8       // waves per block
#define THREADS   (WPB * 32)
#define GE_DIM    35
#define H         16

typedef float v2f __attribute__((ext_vector_type(2)));
typedef float v4f __attribute__((ext_vector_type(4)));
typedef float v8f __attribute__((ext_vector_type(8)));

// Wave-private LDS hand-off: HW keeps same-wave DS ops in order; fence+wave_barrier
// stops the compiler from reordering across the exchange.
__device__ __forceinline__ void wave_lds_sync() {
    __builtin_amdgcn_fence(__ATOMIC_RELEASE, "wavefront");
    __builtin_amdgcn_wave_barrier();
    __builtin_amdgcn_fence(__ATOMIC_ACQUIRE, "wavefront");
}

// first index k in [0,16) with p[k] > 0.5, else 0  (== jnp.argmax of bool)
__device__ __forceinline__ float decode_nib(const float* p) {
    unsigned m = 0;
#pragma unroll
    for (int k = 0; k < 16; ++k) m |= (p[k] > 0.5f) ? (1u << k) : 0u;
    return m ? (float)(__ffs(m) - 1) : 0.0f;
}

__device__ __forceinline__ float mlp_result_scalar(
    float nib, float car, float fa, float fo, float fx,
    const float* __restrict__ W1, const float* __restrict__ b1,
    const float* __restrict__ W2, const float* __restrict__ b2) {
    float s = b2[GE_DIM - 1];
#pragma unroll
    for (int h = 0; h < H; ++h) {
        float z = nib * W1[0 * H + h] + car * W1[1 * H + h]
                + fo  * W1[30 * H + h] + fx * W1[31 * H + h]
                + fa  * W1[32 * H + h] + b1[h];
        s += fmaxf(z, 0.0f) * W2[h * GE_DIM + (GE_DIM - 1)];
    }
    return s;
}

__global__ __launch_bounds__(THREADS) void alu_wmma_kernel(
    const float* __restrict__ x,
    const float* __restrict__ aW1, const float* __restrict__ ab1,
    const float* __restrict__ aW2, const float* __restrict__ ab2,
    const float* __restrict__ oW1, const float* __restrict__ ob1,
    const float* __restrict__ oW2, const float* __restrict__ ob2,
    const float* __restrict__ xW1, const float* __restrict__ xb1,
    const float* __restrict__ xW2, const float* __restrict__ xb2,
    float* __restrict__ out, int nrows)
{
    __shared__ float sh[WPB][RPW * ROWP];
    const int lane  = threadIdx.x & 31;
    const int wave  = threadIdx.x >> 5;
    const int chunk = blockIdx.x * WPB + wave;
    const long base = (long)chunk * (RPW * BD);
    float* smem = sh[wave];

    const bool fullwave = (long)(chunk + 1) * RPW <= (long)nrows;

    if (fullwave) {
        // ---- stage in: 800 floats as 200 aligned float4 (non-temporal stream) ----
        const v4f* in4 = (const v4f*)(x + base);
#pragma unroll
        for (int it = 0; it < 7; ++it) {
            int idx = it * 32 + lane;
            if (idx < 200) {
                v4f v = __builtin_nontemporal_load(&in4[idx]);
                int row = idx / 25, col = (idx % 25) * 4;   // 100 % 4 == 0: no row straddle
                *(v4f*)&smem[row * ROWP + col] = v;
            }
        }
        wave_lds_sync();

        // ---- per-lane decode of pair (lane % 16): pos = pair/2, p = pair&1 ----
        const int pr  = lane & 15;
        const int pos = pr >> 1;
        const int p   = pr & 1;
        const float* row = smem + pos * ROWP;
        const float nibf  = decode_nib(row + 4  + 16 * p);   // ALU_LO/HI
        const float carf  = decode_nib(row + 36 + 16 * p);   // AX_CARRY_LO/HI
        const float markv = row[0];
        const float fa = row[1], fo = row[2], fx = row[3];   // raw flag values

        const bool lo  = lane < 16;
        const int  l16 = lane & 15;

        // ---- A matrices: 16(hidden) x 4(feature), ISA 32-bit A 16x4 layout:
        //      lanes 0-15 hold K=0 (v[0]) / K=1 (v[1]); lanes 16-31 hold K=2 / K=3.
        //      chunk0 features: [nib(W1 row0), car(row1), or(row30), xor(row31)]
        //      chunk1 features: [and(row32),  bias->b1,   0,        0       ]
        v2f A0a, A1a, A0o, A1o, A0x, A1x;
        A0a[0] = lo ? aW1[0 * H + l16]  : aW1[30 * H + l16];
        A0a[1] = lo ? aW1[1 * H + l16]  : aW1[31 * H + l16];
        A1a[0] = lo ? aW1[32 * H + l16] : 0.0f;
        A1a[1] = lo ? ab1[l16]          : 0.0f;
        A0o[0] = lo ? oW1[0 * H + l16]  : oW1[30 * H + l16];
        A0o[1] = lo ? oW1[1 * H + l16]  : oW1[31 * H + l16];
        A1o[0] = lo ? oW1[32 * H + l16] : 0.0f;
        A1o[1] = lo ? ob1[l16]          : 0.0f;
        A0x[0] = lo ? xW1[0 * H + l16]  : xW1[30 * H + l16];
        A0x[1] = lo ? xW1[1 * H + l16]  : xW1[31 * H + l16];
        A1x[0] = lo ? xW1[32 * H + l16] : 0.0f;
        A1x[1] = lo ? xb1[l16]          : 0.0f;

        // ---- W2 RESULT column (per-lane: hidden hb..hb+7), b2[RESULT] ----
        const int hb = lo ? 0 : 8;
        float w2a[8], w2o[8], w2x[8];
#pragma unroll
        for (int v = 0; v < 8; ++v) {
            w2a[v] = aW2[(hb + v) * GE_DIM + (GE_DIM - 1)];
            w2o[v] = oW2[(hb + v) * GE_DIM + (GE_DIM - 1)];
            w2x[v] = xW2[(hb + v) * GE_DIM + (GE_DIM - 1)];
        }
        const float b2a = ab2[GE_DIM - 1], b2o = ob2[GE_DIM - 1], b2x = xb2[GE_DIM - 1];

        // ---- B matrices: 4(feature) x 16(pair); mirrored layout (lanes 0-15: K=0/1,
        //      lanes 16-31: K=2/3, N = lane%16). Every lane already owns its pair's
        //      decoded features -> no shuffles needed to build B.
        v2f B0, B1;
        B0[0] = lo ? nibf : fo;
        B0[1] = lo ? carf : fx;
        B1[0] = lo ? fa   : 0.0f;
        B1[1] = lo ? 1.0f : 0.0f;

        // ---- three tiny MLP hidden layers via V_WMMA_F32_16X16X4_F32 (K=4, x2 chunks) ----
        v8f ca = {}, co = {}, cx = {};
        ca = __builtin_amdgcn_wmma_f32_16x16x4_f32(false, A1a, false, B1, (short)0, ca, false, false);
        ca = __builtin_amdgcn_wmma_f32_16x16x4_f32(false, A0a, false, B0, (short)0, ca, false, false);
        co = __builtin_amdgcn_wmma_f32_16x16x4_f32(false, A1o, false, B1, (short)0, co, false, false);
        co = __builtin_amdgcn_wmma_f32_16x16x4_f32(false, A0o, false, B0, (short)0, co, false, false);
        cx = __builtin_amdgcn_wmma_f32_16x16x4_f32(false, A1x, false, B1, (short)0, cx, false, false);
        cx = __builtin_amdgcn_wmma_f32_16x16x4_f32(false, A0x, false, B0, (short)0, cx, false, false);

        // ---- ReLU + partial dot with W2[:,RESULT] (8 local hidden per lane) ----
        float pa = 0.f, po = 0.f, px = 0.f;
#pragma unroll
        for (int v = 0; v < 8; ++v) {
            pa += fmaxf(ca[v], 0.0f) * w2a[v];
            po += fmaxf(co[v], 0.0f) * w2o[v];
            px += fmaxf(cx[v], 0.0f) * w2x[v];
        }

        // ---- priority select (and > or > xor) on partials, then one xor-16 reduce ----
        const bool ia = fa > 0.5f, io = fo > 0.5f, ix = fx > 0.5f;
        float partial = ia ? pa : (io ? po : px);
        const float b2s = ia ? b2a : (io ? b2o : b2x);
        partial += __shfl_xor(partial, 16, 32);
        const float result = partial + b2s;

        float rr = __builtin_rintf(result);               // round-half-even, like jnp.round
        rr = fminf(fmaxf(rr, 0.0f), 15.0f);
        const int r = (int)rr;
        const bool active = (markv > 0.5f) && (ia | io | ix);

        // lanes 0..15 own distinct (pos,p) -> disjoint OUTPUT blocks, no conflict
        if (lo && active)
            smem[pos * ROWP + 68 + 16 * p + r] += 2.0f;
        wave_lds_sync();

        // ---- stage out: merged rows as non-temporal float4 stream ----
        v4f* out4 = (v4f*)(out + base);
#pragma unroll
        for (int it = 0; it < 7; ++it) {
            int idx = it * 32 + lane;
            if (idx < 200) {
                int row2 = idx / 25, col = (idx % 25) * 4;
                __builtin_nontemporal_store(*(const v4f*)&smem[row2 * ROWP + col], &out4[idx]);
            }
        }
    } else if ((long)chunk * RPW < (long)nrows) {
        // ---- scalar tail (partial wave-chunk): one lane per remaining row ----
        const long r0 = (long)chunk * RPW + lane;
        if (lane < RPW && r0 < (long)nrows) {
            const float* rowg = x + r0 * BD;
            float* rowo = out + r0 * BD;
            for (int i = 0; i < BD; ++i) rowo[i] = rowg[i];
            const float fa = rowg[1], fo = rowg[2], fx = rowg[3];
            const bool ia = fa > 0.5f, io = fo > 0.5f, ix = fx > 0.5f;
            if ((rowg[0] > 0.5f) && (ia | io | ix)) {
                const float* W1 = ia ? aW1 : (io ? oW1 : xW1);
                const float* b1 = ia ? ab1 : (io ? ob1 : xb1);
                const float* W2 = ia ? aW2 : (io ? oW2 : xW2);
                const float* b2 = ia ? ab2 : (io ? ob2 : xb2);
                for (int p = 0; p < 2; ++p) {
                    const float nib = decode_nib(rowg + 4 + 16 * p);
                    const float car = decode_nib(rowg + 36 + 16 * p);
                    const float s = mlp_result_scalar(nib, car, fa, fo, fx, W1, b1, W2, b2);
                    const int rv = (int)fminf(fmaxf(__builtin_rintf(s), 0.0f), 15.0f);
                    rowo[68 + 16 * p + rv] += 2.0f;
                }
            }
        }
    }
}

extern "C" void kernel_launch(void* const* d_in, const int* in_sizes, int n_in,
                              void* d_out, int out_size, void* d_ws, size_t ws_size,
                              hipStream_t stream) {
    (void)n_in; (void)out_size; (void)d_ws; (void)ws_size;
    const float* x = (const float*)d_in[0];
    const float* w[12];
    for (int i = 0; i < 12; ++i) w[i] = (const float*)d_in[1 + i];
    const int nrows  = in_sizes[0] / BD;                 // B*S
    const int chunks = (nrows + RPW - 1) / RPW;          // 8-row wave chunks
    const int blocks = (chunks + WPB - 1) / WPB;
    alu_wmma_kernel<<<blocks, THREADS, 0, stream>>>(
        x,
        w[0], w[1], w[2], w[3],
        w[4], w[5], w[6], w[7],
        w[8], w[9], w[10], w[11],
        (float*)d_out, nrows);
}